// BipartiteGNNConvVariableToFactor_3728031613008
// MI455X (gfx1250) — compile-verified
//
#include <hip/hip_runtime.h>

#define D 128

typedef __attribute__((ext_vector_type(2))) float v2f;
typedef __attribute__((ext_vector_type(4))) float f4;
typedef __attribute__((ext_vector_type(8))) float v8f;

// ---------------------------------------------------------------------------
// P[M x 128] = X[M x 128] @ W[128 x 128]   (W row stride = 128 floats)
// One 256-thread block = 8 waves; wave w owns N-tile [16w, 16w+16).
// B fragments for the full K=128 are preloaded (32 x v2f = 64 VGPRs); each
// wave then processes TWO 16-row M tiles per iteration with independent
// accumulators so the V_WMMA_F32_16X16X4_F32 D->C chains interleave.
// M must be a multiple of 32 (200000 and 100000 both are).
// ---------------------------------------------------------------------------
__global__ void __launch_bounds__(256)
gemm_xw(const float* __restrict__ X, const float* __restrict__ W,
        float* __restrict__ P, int M)
{
    const int lane  = threadIdx.x & 31;
    const int wave  = threadIdx.x >> 5;
    const int nBase = wave * 16;
    const int col   = lane & 15;   // N column within tile (also A row index)
    const int khalf = lane >> 4;   // 0: K pair {0,1}, 1: K pair {2,3}

    // Preload B: lane holds W[k][nBase+col], W[k+1][nBase+col] per K-step.
    v2f bfrag[32];
#pragma unroll
    for (int kk = 0; kk < 32; ++kk) {
        const int k = kk * 4 + khalf * 2;
        bfrag[kk].x = W[(size_t)k * D + nBase + col];
        bfrag[kk].y = W[(size_t)(k + 1) * D + nBase + col];
    }

    const int numTiles = M >> 5;   // 32-row super-tiles
    for (int mt = blockIdx.x; mt < numTiles; mt += gridDim.x) {
        const int mBase = mt << 5;
        const float* arow0 = X + (size_t)(mBase + col) * D + khalf * 2;
        const float* arow1 = arow0 + (size_t)16 * D;
        v8f acc0 = {};
        v8f acc1 = {};
#pragma unroll
        for (int kk = 0; kk < 32; ++kk) {
            v2f a0 = *(const v2f*)(arow0 + kk * 4);
            v2f a1 = *(const v2f*)(arow1 + kk * 4);
            acc0 = __builtin_amdgcn_wmma_f32_16x16x4_f32(
                false, a0, false, bfrag[kk], (short)0, acc0, false, false);
            acc1 = __builtin_amdgcn_wmma_f32_16x16x4_f32(
                false, a1, false, bfrag[kk], (short)0, acc1, false, false);
        }
        // C/D layout: VGPR r -> row r (lanes 0-15) or row r+8 (lanes 16-31)
        float* prow0 = P + (size_t)mBase * D + nBase + col;
        float* prow1 = prow0 + (size_t)16 * D;
#pragma unroll
        for (int r = 0; r < 8; ++r) {
            prow0[(size_t)(r + khalf * 8) * D] = acc0[r];
            prow1[(size_t)(r + khalf * 8) * D] = acc1[r];
        }
    }
}

// ---------------------------------------------------------------------------
// Per-edge message + scatter-sum:
//   aggr[r] += relu(Pf[r] + Pv[s] + ea * w_e + b_msg)
// One wave per edge; each lane handles 4 contiguous floats (b128 loads,
// 4x native fp32 global atomic adds into the L2-resident accumulator).
// ---------------------------------------------------------------------------
__global__ void __launch_bounds__(256)
edge_scatter(const float* __restrict__ Pf, const float* __restrict__ Pv,
             const int* __restrict__ senders, const int* __restrict__ receivers,
             const float* __restrict__ edge_attr,
             const float* __restrict__ w_e, const float* __restrict__ b_msg,
             float* __restrict__ aggr, int E)
{
    const int lane = threadIdx.x & 31;
    const int e    = blockIdx.x * 8 + (threadIdx.x >> 5);
    if (e >= E) return;

    const int   s  = senders[e];
    const int   r  = receivers[e];
    const float ea = edge_attr[e];
    const int   o  = lane * 4;

    f4 pf = *(const f4*)(Pf + (size_t)r * D + o);
    f4 pv = *(const f4*)(Pv + (size_t)s * D + o);
    f4 we = *(const f4*)(w_e + o);
    f4 bm = *(const f4*)(b_msg + o);

    float* dst = aggr + (size_t)r * D + o;
#pragma unroll
    for (int i = 0; i < 4; ++i) {
        float m = pf[i] + pv[i] + ea * we[i] + bm[i];
        m = m > 0.0f ? m : 0.0f;
        float* p = dst + i;
        // Native no-return fp32 atomic add (GLOBAL_ATOMIC_ADD_F32, STOREcnt),
        // device scope: different workgroups accumulate into the same row.
        asm volatile("global_atomic_add_f32 %0, %1, off scope:SCOPE_DEV"
                     :: "v"(p), "v"(m)
                     : "memory");
    }
}

// ---------------------------------------------------------------------------
// out[M x 128] = relu(Fa @ Wc[0:128] + Ag @ Wc[128:256] + bc)
// Same wave/tile mapping, dual M tiles; B streamed from L2 (W_comb = 128 KB,
// cache-hot across all 3125 super-tiles).
// ---------------------------------------------------------------------------
__global__ void __launch_bounds__(256)
combine_gemm(const float* __restrict__ Fa, const float* __restrict__ Ag,
             const float* __restrict__ Wc, const float* __restrict__ bc,
             float* __restrict__ out, int M)
{
    const int lane  = threadIdx.x & 31;
    const int wave  = threadIdx.x >> 5;
    const int nBase = wave * 16;
    const int col   = lane & 15;
    const int khalf = lane >> 4;

    const int numTiles = M >> 5;
    for (int mt = blockIdx.x; mt < numTiles; mt += gridDim.x) {
        const int mBase = mt << 5;
        const float* arowF0 = Fa + (size_t)(mBase + col) * D + khalf * 2;
        const float* arowA0 = Ag + (size_t)(mBase + col) * D + khalf * 2;
        const float* arowF1 = arowF0 + (size_t)16 * D;
        const float* arowA1 = arowA0 + (size_t)16 * D;
        v8f acc0 = {};
        v8f acc1 = {};
#pragma unroll
        for (int kk = 0; kk < 64; ++kk) {
            const int k = kk * 4 + khalf * 2;   // 0..255 over [Fa | Ag]
            v2f b;
            b.x = Wc[(size_t)k * D + nBase + col];
            b.y = Wc[(size_t)(k + 1) * D + nBase + col];
            v2f a0, a1;
            if (kk < 32) {
                a0 = *(const v2f*)(arowF0 + kk * 4);
                a1 = *(const v2f*)(arowF1 + kk * 4);
            } else {
                a0 = *(const v2f*)(arowA0 + (kk - 32) * 4);
                a1 = *(const v2f*)(arowA1 + (kk - 32) * 4);
            }
            acc0 = __builtin_amdgcn_wmma_f32_16x16x4_f32(
                false, a0, false, b, (short)0, acc0, false, false);
            acc1 = __builtin_amdgcn_wmma_f32_16x16x4_f32(
                false, a1, false, b, (short)0, acc1, false, false);
        }
        const float bias = bc[nBase + col];
        float* orow0 = out + (size_t)mBase * D + nBase + col;
        float* orow1 = orow0 + (size_t)16 * D;
#pragma unroll
        for (int r = 0; r < 8; ++r) {
            float v0 = acc0[r] + bias;
            float v1 = acc1[r] + bias;
            orow0[(size_t)(r + khalf * 8) * D] = v0 > 0.0f ? v0 : 0.0f;
            orow1[(size_t)(r + khalf * 8) * D] = v1 > 0.0f ? v1 : 0.0f;
        }
    }
}

// ---------------------------------------------------------------------------
extern "C" void kernel_launch(void* const* d_in, const int* in_sizes, int n_in,
                              void* d_out, int out_size, void* d_ws, size_t ws_size,
                              hipStream_t stream)
{
    const float* variables = (const float*)d_in[0];   // [V,128]
    const float* factors   = (const float*)d_in[1];   // [F,128]
    const int*   senders   = (const int*)  d_in[2];   // [E]
    const int*   receivers = (const int*)  d_in[3];   // [E]
    const float* edge_attr = (const float*)d_in[4];   // [E]
    const float* W_msg     = (const float*)d_in[5];   // [257,128]
    const float* b_msg     = (const float*)d_in[6];   // [128]
    const float* W_comb    = (const float*)d_in[7];   // [256,128]
    const float* b_comb    = (const float*)d_in[8];   // [128]

    const int V = in_sizes[0] / D;
    const int F = in_sizes[1] / D;
    const int E = in_sizes[2];

    // W_msg split: rows [0,128) -> x_i (factors), [128,256) -> x_j (variables),
    // row 256 -> edge-attr weight vector.
    const float* W_i = W_msg;
    const float* W_j = W_msg + (size_t)128 * D;
    const float* w_e = W_msg + (size_t)256 * D;

    // Workspace: Pv [V,128] | Pf [F,128] | aggr [F,128]  (~205 MB total)
    float* Pv   = (float*)d_ws;
    float* Pf   = Pv + (size_t)V * D;
    float* aggr = Pf + (size_t)F * D;

    hipMemsetAsync(aggr, 0, (size_t)F * D * sizeof(float), stream);

    const int tilesV = V >> 5, tilesF = F >> 5;
    const int gV = tilesV < 2048 ? tilesV : 2048;
    const int gF = tilesF < 2048 ? tilesF : 2048;

    gemm_xw<<<gV, 256, 0, stream>>>(variables, W_j, Pv, V);
    gemm_xw<<<gF, 256, 0, stream>>>(factors,   W_i, Pf, F);

    edge_scatter<<<(E + 7) / 8, 256, 0, stream>>>(
        Pf, Pv, senders, receivers, edge_attr, w_e, b_msg, aggr, E);

    combine_gemm<<<gF, 256, 0, stream>>>(
        factors, aggr, W_comb, b_comb, (float*)d_out, F);
}